// CostYoloV2_27891517620437
// MI455X (gfx1250) — compile-verified
//
#include <hip/hip_runtime.h>
#include <hip/hip_bf16.h>

// Problem constants (from reference)
#define BQ   32
#define TQ   50
#define NUMQ 5
#define CLSQ 8
#define ALQ  15       // CLASSES + COORDS + 1 = 8 + 6 + 1
#define HQ   64
#define WQ   64
#define HWQ  4096

typedef __attribute__((ext_vector_type(2))) float v2f;
typedef __attribute__((ext_vector_type(8))) float v8f;

__constant__ float c_AW[NUMQ] = {1.08f, 3.42f, 6.63f, 9.42f, 16.62f};
__constant__ float c_AH[NUMQ] = {1.19f, 4.41f, 11.38f, 5.11f, 10.52f};

__device__ __forceinline__ float sq(float v) { return v * v; }

// ---------------------------------------------------------------------------
// Kernel 1: no-object loss.  One block per (b, anchor) channel-plane group.
// 160 blocks x 256 threads (8 wave32s). Each thread walks 16 cells of the
// 64x64 plane; coalesced streaming loads of 5 contiguous 16KB planes.
// IoU>0.5 test is division-free: inter/union > 0.5  <=>  3*inter > area+tarea.
// Wave reduction done with V_WMMA_F32_16X16X4_F32 (exact fp32).
// ---------------------------------------------------------------------------
__global__ __launch_bounds__(256) void yolo_noobj_kernel(
    const float* __restrict__ x, const float* __restrict__ truth,
    float* __restrict__ partial) {
  __shared__ float s_tl[TQ], s_tr[TQ], s_tu[TQ], s_td[TQ], s_ta[TQ];
  __shared__ float s_red[16];

  const int blk = blockIdx.x;          // 0..159
  const int b = blk / NUMQ;
  const int n = blk % NUMQ;
  const int tid = threadIdx.x;

  // Precompute truth-box bounds for this batch item.
  if (tid < TQ) {
    const float* tp = truth + ((size_t)b * TQ + tid) * 8;
    const float tcx = tp[0], tcy = tp[1], tw = tp[2], th = tp[3];
    s_tl[tid] = tcx - 0.5f * tw;
    s_tr[tid] = tcx + 0.5f * tw;
    s_tu[tid] = tcy - 0.5f * th;
    s_td[tid] = tcy + 0.5f * th;
    s_ta[tid] = tw * th;
  }
  __syncthreads();

  const float awn = c_AW[n] * (1.0f / WQ);
  const float ahn = c_AH[n] * (1.0f / HQ);
  const float* p0 = x + (size_t)(b * NUMQ * ALQ + n * ALQ) * HWQ;  // tx plane
  const float* p1 = p0 + 1 * HWQ;                                  // ty
  const float* p2 = p0 + 2 * HWQ;                                  // tw
  const float* p3 = p0 + 3 * HWQ;                                  // th
  const float* p6 = p0 + 6 * HWQ;                                  // obj

  float acc = 0.0f;
#pragma unroll 1
  for (int iter = 0; iter < 16; ++iter) {
    const int cell = tid + iter * 256;
    if (iter < 15) {  // uniform branch: prefetch next tile (global_prefetch_b8)
      __builtin_prefetch(p0 + cell + 256, 0, 0);
      __builtin_prefetch(p2 + cell + 256, 0, 0);
      __builtin_prefetch(p6 + cell + 256, 0, 0);
    }
    const float tx = p0[cell];
    const float ty = p1[cell];
    const float tw = p2[cell];
    const float th = p3[cell];
    const float obj = p6[cell];

    const float cx = (tx + (float)(cell & (WQ - 1))) * (1.0f / WQ);
    const float cy = (ty + (float)(cell >> 6)) * (1.0f / HQ);
    const float pw = expf(tw) * awn;
    const float ph = expf(th) * ahn;
    const float hx = 0.5f * pw, hy = 0.5f * ph;
    const float x0 = cx - hx, x1 = cx + hx;
    const float y0 = cy - hy, y1 = cy + hy;
    const float area = pw * ph;

    bool masked = false;
    for (int t = 0; t < TQ; ++t) {
      const float l = fmaxf(x0, s_tl[t]);
      const float r = fminf(x1, s_tr[t]);
      const float u = fmaxf(y0, s_tu[t]);
      const float d = fminf(y1, s_td[t]);
      const float iw = fmaxf(r - l, 0.0f);
      const float ih = fmaxf(d - u, 0.0f);
      const float inter = iw * ih;
      masked = masked || (3.0f * inter > area + s_ta[t]);  // iou > 0.5
    }
    acc = masked ? acc : acc + obj * obj;  // (obj - obj*mask)^2
  }

  // --- exact fp32 wave reduction via V_WMMA_F32_16X16X4_F32 ------------------
  // A[i][0]=acc(lane i), A[i][1]=0, A[i][2]=acc(lane 16+i), A[i][3]=0;
  // B = ones  =>  D[i][j] = acc[i] + acc[i+16].  Each lane then sums its 8 D
  // registers: lanes 0-15 hold sum(acc[0..7],acc[16..23]), lanes 16-31 the rest.
  v2f a;     a.x = acc;  a.y = 0.0f;
  v2f ones;  ones.x = 1.0f; ones.y = 1.0f;
  v8f cmat = {};
  v8f dmat = __builtin_amdgcn_wmma_f32_16x16x4_f32(
      /*neg_a=*/false, a, /*neg_b=*/false, ones,
      /*c_mod=*/(short)0, cmat, /*reuse_a=*/false, /*reuse_b=*/false);
  float s = dmat[0] + dmat[1] + dmat[2] + dmat[3] +
            dmat[4] + dmat[5] + dmat[6] + dmat[7];

  const int lane = tid & 31;
  const int wv = tid >> 5;
  if (lane == 0)  s_red[wv * 2 + 0] = s;
  if (lane == 16) s_red[wv * 2 + 1] = s;
  __syncthreads();
  if (tid == 0) {
    float tot = 0.0f;
    for (int i = 0; i < 16; ++i) tot += s_red[i];  // fixed order: deterministic
    partial[blk] = tot;
  }
}

// ---------------------------------------------------------------------------
// Kernel 2: per-truth losses. 32 blocks (one per batch item) x 64 threads
// (2 wave32s); thread t handles truth box t. Tiny gather workload.
// ---------------------------------------------------------------------------
__global__ __launch_bounds__(64) void yolo_truth_kernel(
    const float* __restrict__ x, const float* __restrict__ truth,
    float* __restrict__ partial) {
  __shared__ float red[64];
  const int b = blockIdx.x;
  const int t = threadIdx.x;

  float loss = 0.0f;
  if (t < TQ) {
    const float* tp = truth + ((size_t)b * TQ + t) * 8;
    const float t_cx = tp[0], t_cy = tp[1], t_w = tp[2], t_h = tp[3];
    const bool valid = (t_w > 0.0f) && (t_h > 0.0f);

    int gi = (int)(t_cx * (float)WQ); gi = gi < 0 ? 0 : (gi > WQ - 1 ? WQ - 1 : gi);
    int gj = (int)(t_cy * (float)HQ); gj = gj < 0 ? 0 : (gj > HQ - 1 ? HQ - 1 : gj);

    const size_t cellbase = (size_t)b * NUMQ * ALQ * HWQ + (size_t)gj * WQ + gi;

    // Pick the anchor whose (rw,rh) box best matches (t_w,t_h); argmax w/
    // strict > keeps jnp.argmax first-max tie semantics.
    float best_iou = -1.0f;
    int best_n = 0;
    for (int n = 0; n < NUMQ; ++n) {
      const float cw = x[cellbase + (size_t)(n * ALQ + 2) * HWQ];
      const float ch = x[cellbase + (size_t)(n * ALQ + 3) * HWQ];
      const float rw = expf(cw) * c_AW[n] * (1.0f / WQ);
      const float rh = expf(ch) * c_AH[n] * (1.0f / HQ);
      const float inter2 = fminf(rw, t_w) * fminf(rh, t_h);
      const float union2 = rw * rh + t_w * t_h - inter2;
      const float iou = inter2 / union2;
      if (iou > best_iou) { best_iou = iou; best_n = n; }
    }

    float bc[ALQ];
    for (int c = 0; c < ALQ; ++c)
      bc[c] = x[cellbase + (size_t)(best_n * ALQ + c) * HWQ];

    const float aw_b = c_AW[best_n], ah_b = c_AH[best_n];
    const float safe_w = valid ? t_w : 1.0f;
    const float safe_h = valid ? t_h : 1.0f;
    const float bt0 = t_cx * (float)WQ - (float)gi;
    const float bt1 = t_cy * (float)HQ - (float)gj;
    const float bt2 = logf(safe_w * (float)WQ / aw_b);
    const float bt3 = logf(safe_h * (float)HQ / ah_b);

    float lc = sq(bc[0] - bt0) + sq(bc[1] - bt1) +
               sq(bc[2] - bt2) + sq(bc[3] - bt3);          // COORD_SCALE = 1
    float lf = sq(bc[4] - tp[5]) + sq(bc[5] - tp[6]);
    float lo = 5.0f * sq(bc[6] - 1.0f);                     // OBJECT_SCALE = 5
    const int cls = (int)tp[7];
    float lcl = 0.0f;
    for (int c = 0; c < CLSQ; ++c) {
      const float tgt = (c == cls) ? 1.0f : 0.0f;
      lcl += sq(bc[7 + c] - tgt);                           // CLASS_SCALE = 1
    }
    loss = valid ? (lc + lf + lo + lcl) : 0.0f;
  }

  red[t] = loss;
  __syncthreads();
  if (t == 0) {
    float s = 0.0f;
    for (int i = 0; i < 64; ++i) s += red[i];  // fixed order: deterministic
    partial[b] = s;
  }
}

// ---------------------------------------------------------------------------
// Kernel 3: deterministic serial finalize of 160 + 32 partials into d_out[0].
// ---------------------------------------------------------------------------
__global__ void yolo_finalize_kernel(const float* __restrict__ partial,
                                     float* __restrict__ out) {
  if (threadIdx.x == 0 && blockIdx.x == 0) {
    float s = 0.0f;
    for (int i = 0; i < BQ * NUMQ + BQ; ++i) s += partial[i];
    out[0] = s;
  }
}

extern "C" void kernel_launch(void* const* d_in, const int* in_sizes, int n_in,
                              void* d_out, int out_size, void* d_ws, size_t ws_size,
                              hipStream_t stream) {
  const float* x = (const float*)d_in[0];      // (32, 75, 64, 64) f32
  const float* truth = (const float*)d_in[1];  // (32, 50, 8) f32
  float* out = (float*)d_out;                  // scalar f32
  float* partial = (float*)d_ws;               // 192 floats of scratch

  yolo_noobj_kernel<<<BQ * NUMQ, 256, 0, stream>>>(x, truth, partial);
  yolo_truth_kernel<<<BQ, 64, 0, stream>>>(x, truth, partial + BQ * NUMQ);
  yolo_finalize_kernel<<<1, 32, 0, stream>>>(partial, out);
}